// ClementsBell_Arct_82042465288888
// MI455X (gfx1250) — compile-verified
//
#include <hip/hip_runtime.h>
#include <math.h>

// ---------------------------------------------------------------------------
// Clements mesh forward: M = PC_out * (prod of 256 steps) * PC_in * I
// Column-parallel: each wave32 owns one of the 512 columns of M (in LDS).
// Per step, the six thin sublayers are fused into two 2x2 complex block ops:
//   G_even[p] = E(le1) * P(th0) * E(le0)   on rows (2p, 2p+1)
//   G_odd[q]  = O(lo1) * P(th1) * O(lo0)   on rows (2q+1, 2q+2), edge rows
//               0 and 511 receive only the P(th1) phase.
// The G matrices depend only on (step, pair) — identical for all columns —
// so the block's 128 threads precompute them once per step into LDS
// (4x transcendental amortization, 4-sqrt fused radicals); waves then apply
// pure FMA + b128 LDS ops. Raw per-step parameters (~16KB) are prefetched
// with CDNA5 async global->LDS loads (ASYNCcnt) overlapping the apply phase.
// ---------------------------------------------------------------------------

#define NDIM 512
#define HDIM 256
#define WAVES 4                 // columns (waves) per block
#define TPB   (WAVES * 32)

// raw parameter LDS layout (float offsets), 4096 floats = 16KB
#define OFF_TH 0                // 2 x 512 thetas_full rows (layers 2i, 2i+1)
#define OFF_LP 1024             // 2 x 512 pc_losses_full rows
#define OFF_LE 2048             // 2 x 256 mmi_losses_even rows
#define OFF_IE 2560             // 2 x 256 mmi_imb_even rows
#define OFF_LO 3072             // 2 x 255 mmi_losses_odd rows (stride-256 slots)
#define OFF_IO 3584             // 2 x 255 mmi_imb_odd rows  (stride-256 slots)
#define PBUF   4096

// composed-constant LDS layout (float offsets)
#define C_EV   0                // 256 pairs x 8 floats (two float4 per pair)
#define C_OD   2048             // 255 pairs x 8 floats (slot 255 unused)
#define C_EDGE 4096             // c(row0), c(row511) of pc layer 2i+1
#define CBUF   4104

#ifndef __has_builtin
#define __has_builtin(x) 0
#endif

#if __has_builtin(__builtin_amdgcn_global_load_async_to_lds_b128) && \
    __has_builtin(__builtin_amdgcn_global_load_async_to_lds_b32)
#define HAVE_ASYNC 1
#else
#define HAVE_ASYNC 0
#endif

typedef __attribute__((ext_vector_type(4))) int v4i;
#define GAS1_V4 __attribute__((address_space(1))) v4i*
#define LAS3_V4 __attribute__((address_space(3))) v4i*
#define GAS1_I  __attribute__((address_space(1))) int*
#define LAS3_I  __attribute__((address_space(3))) int*

#if HAVE_ASYNC
// args: (global src, lds dst, imm offset, imm cpol)
#define ASYNC_CP4(dst, src)                                            \
  __builtin_amdgcn_global_load_async_to_lds_b128(                      \
      (GAS1_V4)(src), (LAS3_V4)(dst), 0, 0)
#define ASYNC_CP1(dst, src)                                            \
  __builtin_amdgcn_global_load_async_to_lds_b32(                       \
      (GAS1_I)(src), (LAS3_I)(dst), 0, 0)
#else
#define ASYNC_CP4(dst, src) (*(dst) = *(src))
#define ASYNC_CP1(dst, src) (*(dst) = *(src))
#endif

__device__ __forceinline__ void async_wait_all() {
#if HAVE_ASYNC
#if __has_builtin(__builtin_amdgcn_s_wait_asynccnt)
  __builtin_amdgcn_s_wait_asynccnt(0);
#else
  asm volatile("s_wait_asynccnt 0" ::: "memory");
#endif
#endif
}

// Raw v_sqrt_f32: all sqrt inputs lie in [0.2, 1.0], far from denormals.
__device__ __forceinline__ float fsqrt_fast(float x) {
#if __has_builtin(__builtin_amdgcn_sqrtf)
  return __builtin_amdgcn_sqrtf(x);
#else
  return sqrtf(x);
#endif
}

// Intra-wave ordering only (LDS ops of one wave complete in order); just stop
// the compiler from reordering LDS accesses across cross-lane hand-offs.
__device__ __forceinline__ void wavebar() {
  __builtin_amdgcn_wave_barrier();
  asm volatile("" ::: "memory");
}

// ---- constant builders ----------------------------------------------------

__device__ __forceinline__ float2 pc_c(float theta, float loss) {
  float a = fsqrt_fast(1.0f - loss);
  float sn, cs;
  __sincosf(theta, &sn, &cs);
  return make_float2(a * cs, a * sn);
}

// G = E(t2,r2) * diag(c0,c1) * E(t1,r1), E = [[t, i r],[i r, t]]
// G00 = A c0 - B c1 ; G01 = i(C c0 + D c1) ; G10 = i(D c0 + C c1) ;
// G11 = A c1 - B c0 ; with A=t1t2, B=r1r2, C=t2r1, D=r2t1.
// A..D computed with fused radicals: t_k = sqrt((1-l_k)(0.5+i_k)),
// r_k = sqrt((1-l_k)(0.5-i_k)); all factors positive -> products commute
// under sqrt, so 4 sqrts suffice for the 4 coefficients.
__device__ __forceinline__ void make_G(float l1, float i1, float l2, float i2,
                                       float2 c0, float2 c1,
                                       float4* g0, float4* g1) {
  float P   = (1.0f - l1) * (1.0f - l2);
  float up  = 0.5f + i1, un = 0.5f - i1;
  float vp  = 0.5f + i2, vn = 0.5f - i2;
  float Pup = P * up, Pun = P * un;
  float A = fsqrt_fast(Pup * vp);   // t1*t2
  float B = fsqrt_fast(Pun * vn);   // r1*r2
  float C = fsqrt_fast(Pun * vp);   // t2*r1
  float D = fsqrt_fast(Pup * vn);   // r2*t1
  float g00r = A * c0.x - B * c1.x, g00i = A * c0.y - B * c1.y;
  float e01r = C * c0.x + D * c1.x, e01i = C * c0.y + D * c1.y;
  float e10r = D * c0.x + C * c1.x, e10i = D * c0.y + C * c1.y;
  float g11r = A * c1.x - B * c0.x, g11i = A * c1.y - B * c0.y;
  *g0 = make_float4(g00r, g00i, -e01i, e01r);   // [G00 | G01]
  *g1 = make_float4(-e10i, e10r, g11r, g11i);   // [G10 | G11]
}

// ---- per-step constant precompute (all 128 threads cooperate) -------------

__device__ __forceinline__ void precompute_consts(const float* pb, float* cb,
                                                  int tid) {
  float4* CEV = (float4*)(cb + C_EV);
  float4* COD = (float4*)(cb + C_OD);
  const float2* th0 = (const float2*)(pb + OFF_TH);
  const float2* lp0 = (const float2*)(pb + OFF_LP);

#pragma unroll
  for (int u = 0; u < 2; ++u) {                 // even pairs: 256 / 128
    int k = tid + (u << 7);
    float2 t2 = th0[k], l2 = lp0[k];
    float2 c0 = pc_c(t2.x, l2.x);
    float2 c1 = pc_c(t2.y, l2.y);
    float4 g0, g1;
    make_G(pb[OFF_LE + k],        pb[OFF_IE + k],
           pb[OFF_LE + HDIM + k], pb[OFF_IE + HDIM + k],
           c0, c1, &g0, &g1);
    CEV[2 * k]     = g0;
    CEV[2 * k + 1] = g1;
  }
#pragma unroll
  for (int u = 0; u < 2; ++u) {                 // odd pairs: 255 / 128
    int k = tid + (u << 7);
    if (k < HDIM - 1) {
      float2 c0 = pc_c(pb[OFF_TH + NDIM + 2 * k + 1],
                       pb[OFF_LP + NDIM + 2 * k + 1]);
      float2 c1 = pc_c(pb[OFF_TH + NDIM + 2 * k + 2],
                       pb[OFF_LP + NDIM + 2 * k + 2]);
      float4 g0, g1;
      make_G(pb[OFF_LO + k],        pb[OFF_IO + k],
             pb[OFF_LO + HDIM + k], pb[OFF_IO + HDIM + k],
             c0, c1, &g0, &g1);
      COD[2 * k]     = g0;
      COD[2 * k + 1] = g1;
    }
  }
  if (tid == 0) {                               // edge rows get P(th1) only
    float2 c = pc_c(pb[OFF_TH + NDIM], pb[OFF_LP + NDIM]);
    cb[C_EDGE] = c.x; cb[C_EDGE + 1] = c.y;
  }
  if (tid == 1) {
    float2 c = pc_c(pb[OFF_TH + NDIM + NDIM - 1],
                    pb[OFF_LP + NDIM + NDIM - 1]);
    cb[C_EDGE + 2] = c.x; cb[C_EDGE + 3] = c.y;
  }
}

// ---- fused sublayer application (one wave = one column) -------------------
// M column as float4[256]: element p = {re(2p), im(2p), re(2p+1), im(2p+1)}

__device__ __forceinline__ void apply_evenG(float4* M4, const float4* CEV,
                                            int lane) {
#pragma unroll
  for (int s = 0; s < 8; ++s) {
    int p = lane + (s << 5);
    float4 g0 = CEV[2 * p], g1 = CEV[2 * p + 1];
    float4 v = M4[p];
    M4[p] = make_float4(
        g0.x * v.x - g0.y * v.y + g0.z * v.z - g0.w * v.w,
        g0.x * v.y + g0.y * v.x + g0.z * v.w + g0.w * v.z,
        g1.x * v.x - g1.y * v.y + g1.z * v.z - g1.w * v.w,
        g1.x * v.y + g1.y * v.x + g1.z * v.w + g1.w * v.z);
  }
}

__device__ __forceinline__ void apply_oddG(float2* Mc, const float4* COD,
                                           const float* edge, int lane) {
#pragma unroll
  for (int s = 0; s < 8; ++s) {
    int q = lane + (s << 5);
    if (q < HDIM - 1) {
      float4 g0 = COD[2 * q], g1 = COD[2 * q + 1];
      float2 a = Mc[2 * q + 1], b = Mc[2 * q + 2];
      Mc[2 * q + 1] = make_float2(
          g0.x * a.x - g0.y * a.y + g0.z * b.x - g0.w * b.y,
          g0.x * a.y + g0.y * a.x + g0.z * b.y + g0.w * b.x);
      Mc[2 * q + 2] = make_float2(
          g1.x * a.x - g1.y * a.y + g1.z * b.x - g1.w * b.y,
          g1.x * a.y + g1.y * a.x + g1.z * b.y + g1.w * b.x);
    }
  }
  if (lane == 0) {
    float cr = edge[0], ci = edge[1];
    float2 v = Mc[0];
    Mc[0] = make_float2(v.x * cr - v.y * ci, v.x * ci + v.y * cr);
  }
  if (lane == 31) {
    float cr = edge[2], ci = edge[3];
    float2 v = Mc[NDIM - 1];
    Mc[NDIM - 1] = make_float2(v.x * cr - v.y * ci, v.x * ci + v.y * cr);
  }
}

// ---- raw parameter staging into LDS (async, all 128 threads) --------------

__device__ __forceinline__ void stage_step(
    int i, float* pb,
    const float* th_g, const float* lp_g,
    const float* le_g, const float* ie_g,
    const float* lo_g, const float* io_g, int tid) {
  const float4* sth = (const float4*)(th_g + (size_t)(2 * i) * NDIM);
  const float4* slp = (const float4*)(lp_g + (size_t)(2 * i) * NDIM);
  const float4* sle = (const float4*)(le_g + (size_t)(2 * i) * HDIM);
  const float4* sie = (const float4*)(ie_g + (size_t)(2 * i) * HDIM);
  float4* dth = (float4*)(pb + OFF_TH);
  float4* dlp = (float4*)(pb + OFF_LP);
  float4* dle = (float4*)(pb + OFF_LE);
  float4* die = (float4*)(pb + OFF_IE);

  ASYNC_CP4(dth + tid,       sth + tid);        // 2x512 thetas
  ASYNC_CP4(dth + tid + 128, sth + tid + 128);
  ASYNC_CP4(dlp + tid,       slp + tid);        // 2x512 pc losses
  ASYNC_CP4(dlp + tid + 128, slp + tid + 128);
  ASYNC_CP4(dle + tid,       sle + tid);        // 2x256 even losses
  ASYNC_CP4(die + tid,       sie + tid);        // 2x256 even imbalance

  // odd arrays: rows are 255 long (not 16B-row-aligned) -> b32 async copies
  const float* slo = lo_g + (size_t)(2 * i) * (HDIM - 1);
  const float* sio = io_g + (size_t)(2 * i) * (HDIM - 1);
#pragma unroll
  for (int k = 0; k < 4; ++k) {
    int idx = tid + (k << 7);                  // 0..511, need 0..509
    if (idx < 2 * (HDIM - 1)) {
      int sub = (idx >= (HDIM - 1)) ? 1 : 0;
      int q = idx - sub * (HDIM - 1);
      ASYNC_CP1(pb + OFF_LO + sub * HDIM + q, slo + idx);
      ASYNC_CP1(pb + OFF_IO + sub * HDIM + q, sio + idx);
    }
  }
}

// ---------------------------------------------------------------------------

__global__ __launch_bounds__(TPB) void ClementsBell_Arct_82042465288888_kernel(
    const float* __restrict__ thetas_full,   // (512,512)
    const float* __restrict__ thetas_inout,  // (2,512)
    const float* __restrict__ pc_full,       // (512,512)
    const float* __restrict__ pc_inout,      // (2,512)
    const float* __restrict__ le_g,          // (512,256)
    const float* __restrict__ lo_g,          // (512,255)
    const float* __restrict__ ie_g,          // (512,256)
    const float* __restrict__ io_g,          // (512,255)
    float2* __restrict__ out)                // (512,512) complex64
{
  __shared__ __align__(16) float2 Mcol[WAVES][NDIM];   // 16 KB
  __shared__ __align__(16) float praw[PBUF];           // 16 KB raw params
  __shared__ __align__(16) float cbuf[CBUF];           // 16 KB fused consts

  const int tid  = threadIdx.x;
  const int lane = tid & 31;
  const int wv   = tid >> 5;
  const int col  = blockIdx.x * WAVES + wv;
  float2* Mc = Mcol[wv];
  float4* M4 = (float4*)Mc;
  const float4* CEV = (const float4*)(cbuf + C_EV);
  const float4* COD = (const float4*)(cbuf + C_OD);

  // M = I, then input PC: column `col` is e_col * a0[col]
#pragma unroll
  for (int s = 0; s < 8; ++s)
    M4[lane + (s << 5)] = make_float4(0.f, 0.f, 0.f, 0.f);
  if (lane == (col & 31)) {
    Mc[col] = pc_c(thetas_inout[col], pc_inout[col]);
  }

  // stage step 0 raw params
  stage_step(0, praw, thetas_full, pc_full, le_g, ie_g, lo_g, io_g, tid);
  async_wait_all();
  __syncthreads();

  for (int i = 0; i < HDIM; ++i) {
    precompute_consts(praw, cbuf, tid);             // consts for step i
    __syncthreads();                                // consts visible; raw free

    if (i + 1 < HDIM)   // async prefetch next step's raw params (overwrites)
      stage_step(i + 1, praw, thetas_full, pc_full, le_g, ie_g, lo_g, io_g,
                 tid);

    apply_evenG(M4, CEV, lane);                     // E*P*E on even pairs
    wavebar();
    apply_oddG(Mc, COD, cbuf + C_EDGE, lane);       // O*P*O on odd pairs
    wavebar();

    async_wait_all();     // next-step raw staged (issued before apply)
    __syncthreads();      // all waves done with consts before overwrite
  }

  // output PC + store column `col` of M (row-major complex64)
  const float2* thio2 = (const float2*)(thetas_inout + NDIM);
  const float2* pcio2 = (const float2*)(pc_inout + NDIM);
#pragma unroll
  for (int s = 0; s < 8; ++s) {
    int p = lane + (s << 5);                 // row pair (2p, 2p+1)
    float2 t2 = thio2[p], l2 = pcio2[p];
    float2 c0 = pc_c(t2.x, l2.x);
    float2 c1 = pc_c(t2.y, l2.y);
    float4 v = M4[p];
    out[(size_t)(2 * p) * NDIM + col] =
        make_float2(v.x * c0.x - v.y * c0.y, v.x * c0.y + v.y * c0.x);
    out[(size_t)(2 * p + 1) * NDIM + col] =
        make_float2(v.z * c1.x - v.w * c1.y, v.z * c1.y + v.w * c1.x);
  }
}

extern "C" void kernel_launch(void* const* d_in, const int* in_sizes, int n_in,
                              void* d_out, int out_size, void* d_ws,
                              size_t ws_size, hipStream_t stream) {
  (void)in_sizes; (void)n_in; (void)out_size; (void)d_ws; (void)ws_size;
  const float* thetas_full  = (const float*)d_in[0];
  const float* thetas_inout = (const float*)d_in[1];
  const float* pc_full      = (const float*)d_in[2];
  const float* pc_inout     = (const float*)d_in[3];
  const float* mmi_le       = (const float*)d_in[4];
  const float* mmi_lo       = (const float*)d_in[5];
  const float* mmi_ie       = (const float*)d_in[6];
  const float* mmi_io       = (const float*)d_in[7];

  dim3 grid(NDIM / WAVES);   // 128 blocks
  dim3 block(TPB);           // 128 threads = 4 wave32s
  ClementsBell_Arct_82042465288888_kernel<<<grid, block, 0, stream>>>(
      thetas_full, thetas_inout, pc_full, pc_inout, mmi_le, mmi_lo, mmi_ie,
      mmi_io, (float2*)d_out);
}